// Voxelization_78700980732085
// MI455X (gfx1250) — compile-verified
//
#include <hip/hip_runtime.h>
#include <hip/hip_bf16.h>
#include <stdint.h>

#define B_ 8
#define N_ 32768
#define C_ 64
#define R_ 32
#define R3_ (R_ * R_ * R_)
#define TPB 256
#define PTILE 128          // points per block tile
#define NSLOT 64           // LDS hash slots (power of 2)

typedef unsigned int u32x4 __attribute__((ext_vector_type(4)));
typedef int          i32x4 __attribute__((ext_vector_type(4)));
typedef int          i32x8 __attribute__((ext_vector_type(8)));

// ---------------------------------------------------------------------------
// TDM: async DMA of a (PTILE x 64) f32 tile from global memory into LDS.
// Descriptor built per CDNA5 ISA D# layout (group0 128b, group1 256b).
// ---------------------------------------------------------------------------
__device__ __forceinline__ void tdm_load_tile(const float* gptr, uint32_t lds_off) {
    uint64_t ga = (uint64_t)(uintptr_t)gptr;
    u32x4 g0;
    g0.x = 1u;                                            // count=1 (valid user D#)
    g0.y = lds_off;                                       // lds_addr (bytes)
    g0.z = (uint32_t)(ga & 0xFFFFFFFFull);                // global_addr[31:0]
    g0.w = (uint32_t)((ga >> 32) & 0x1FFFFFFull)          // global_addr[56:32]
         | (2u << 30);                                    // type = 2 ("image")
    i32x8 g1;
    g1[0] = (int)(2u << 16);                              // data_size=4B, mask=0
    g1[1] = (int)((uint32_t)C_ << 16);                    // tensor_dim0 lo16 = 64
    g1[2] = (int)((uint32_t)PTILE << 16);                 // dim0 hi=0 | tensor_dim1 lo16 = 128
    g1[3] = (int)((uint32_t)C_ << 16);                    // dim1 hi=0 | tile_dim0 = 64
    g1[4] = (int)(uint32_t)PTILE;                         // tile_dim1 = 128, tile_dim2 = 0
    g1[5] = (int)(uint32_t)C_;                            // tensor_dim0_stride lo32 = 64
    g1[6] = 0;                                            // stride0 hi | stride1 lo
    g1[7] = 0;                                            // stride1 hi
    i32x4 gz = {0, 0, 0, 0};
#if defined(__clang_major__) && (__clang_major__ >= 23)
    i32x8 gz8 = {0, 0, 0, 0, 0, 0, 0, 0};
    __builtin_amdgcn_tensor_load_to_lds(g0, g1, gz, gz, gz8, 0);
#else
    __builtin_amdgcn_tensor_load_to_lds(g0, g1, gz, gz, 0);
#endif
}

// ---------------------------------------------------------------------------
// Pass 1: per-batch sum of squares of centered coords (per axis), for denom.
// Wave32 shuffle reduction, then 3 atomics per wave. Block never straddles a
// batch (N % TPB == 0).
// ---------------------------------------------------------------------------
__global__ void k_sumsq(const float* __restrict__ coords, float* __restrict__ sumsq) {
    int g = blockIdx.x * blockDim.x + threadIdx.x;      // [0, B*N)
    int b = g / N_;
    const float* c = coords + (size_t)g * 3;
    float x = c[0], y = c[1], z = c[2];
    float m = (x + y + z) / 3.0f;                        // mean over xyz (dim 2)
    float dx = x - m, dy = y - m, dz = z - m;
    float sx = dx * dx, sy = dy * dy, sz = dz * dz;
    for (int off = 16; off > 0; off >>= 1) {
        sx += __shfl_down(sx, off, 32);
        sy += __shfl_down(sy, off, 32);
        sz += __shfl_down(sz, off, 32);
    }
    if ((threadIdx.x & 31) == 0) {
        atomicAdd(&sumsq[b * 3 + 0], sx);
        atomicAdd(&sumsq[b * 3 + 1], sy);
        atomicAdd(&sumsq[b * 3 + 2], sz);
    }
}

// ---------------------------------------------------------------------------
// Pass 2: normalize coords, write nc, and scatter-accumulate features into the
// voxel grid. Per-block LDS hash pre-aggregation (ds_add_f32) with TDM-staged
// feature tile; flush unique slots to global with f32 atomics.
// ---------------------------------------------------------------------------
__global__ void k_voxelize(const float* __restrict__ features,
                           const float* __restrict__ coords,
                           const float* __restrict__ sumsq,
                           float* __restrict__ vox,      // (B, C, R3), pre-zeroed
                           float* __restrict__ nc_out,   // (B, N, 3)
                           float* __restrict__ counts)   // (B, R3), pre-zeroed
{
    __shared__ float feat[PTILE * C_];                   // 32 KB TDM landing tile
    __shared__ float sums[NSLOT][C_ + 1];                // [..][64] = count
    __shared__ int   keys[NSLOT];
    __shared__ int   slotof[PTILE];
    __shared__ float denom_sh;

    const int tid       = threadIdx.x;
    const int tilesPerB = N_ / PTILE;
    const int b         = blockIdx.x / tilesPerB;
    const int n0        = (blockIdx.x % tilesPerB) * PTILE;

    for (int i = tid; i < NSLOT; i += TPB) keys[i] = -1;
    for (int i = tid; i < NSLOT * (C_ + 1); i += TPB) ((float*)sums)[i] = 0.0f;

    if (tid == 0) {
        float s0 = sumsq[b * 3 + 0], s1 = sumsq[b * 3 + 1], s2 = sumsq[b * 3 + 2];
        denom_sh = 2.0f * sqrtf(fmaxf(s0, fmaxf(s1, s2)));
        // Async tensor DMA of this block's feature tile into LDS.
        tdm_load_tile(features + ((size_t)b * N_ + n0) * C_,
                      (uint32_t)(uintptr_t)feat);
        __builtin_amdgcn_s_wait_tensorcnt(0);
    }
    __syncthreads();

    // Phase A: per-point normalize + hash-insert voxel index.
    if (tid < PTILE) {
        const float* cp = coords + ((size_t)b * N_ + n0 + tid) * 3;
        float x = cp[0], y = cp[1], z = cp[2];
        float m = (x + y + z) / 3.0f;
        float d = denom_sh;
        float nx = fminf(fmaxf(((x - m) / d + 0.5f) * (float)R_, 0.0f), (float)(R_ - 1));
        float ny = fminf(fmaxf(((y - m) / d + 0.5f) * (float)R_, 0.0f), (float)(R_ - 1));
        float nz = fminf(fmaxf(((z - m) / d + 0.5f) * (float)R_, 0.0f), (float)(R_ - 1));
        float* np = nc_out + ((size_t)b * N_ + n0 + tid) * 3;
        np[0] = nx; np[1] = ny; np[2] = nz;
        int ix = (int)rintf(nx), iy = (int)rintf(ny), iz = (int)rintf(nz); // RTE = jnp.round
        int idx = ix * (R_ * R_) + iy * R_ + iz;

        int slot  = (int)((((uint32_t)idx * 0x9E3779B1u) >> 26) & (NSLOT - 1));
        int found = -1;
        for (int probe = 0; probe < NSLOT; ++probe) {
            int prev = atomicCAS(&keys[slot], -1, idx);  // ds_cmpstore_rtn_b32
            if (prev == -1 || prev == idx) { found = slot; break; }
            slot = (slot + 1) & (NSLOT - 1);
        }
        if (found >= 0) {
            slotof[tid] = found;
            atomicAdd(&sums[found][C_], 1.0f);           // ds_add_f32
        } else {                                         // hash full: direct path
            slotof[tid] = -2 - idx;
            atomicAdd(&counts[(size_t)b * R3_ + idx], 1.0f);
        }
    }
    __syncthreads();

    // Phase B: accumulate features from the TDM tile into LDS slots.
    for (int w = tid; w < PTILE * C_; w += TPB) {
        int p = w >> 6, c = w & (C_ - 1);
        float f = feat[w];
        int s = slotof[p];
        if (s >= 0) {
            atomicAdd(&sums[s][c], f);                   // ds_add_f32
        } else {
            int idx = -2 - s;
            atomicAdd(&vox[((size_t)b * C_ + c) * R3_ + idx], f);
        }
    }
    __syncthreads();

    // Flush: one global atomic per (occupied slot, channel).
    for (int w = tid; w < NSLOT * C_; w += TPB) {
        int s = w >> 6, c = w & (C_ - 1);
        int k = keys[s];
        if (k >= 0) {
            float v = sums[s][c];
            if (v != 0.0f)
                atomicAdd(&vox[((size_t)b * C_ + c) * R3_ + k], v);
        }
    }
    for (int s = tid; s < NSLOT; s += TPB) {
        int k = keys[s];
        if (k >= 0)
            atomicAdd(&counts[(size_t)b * R3_ + k], sums[s][C_]);
    }
}

// ---------------------------------------------------------------------------
// Pass 3: divide sums by max(count, 1). Voxel index fastest-varying =>
// fully coalesced.
// ---------------------------------------------------------------------------
__global__ void k_div(float* __restrict__ vox, const float* __restrict__ counts) {
    size_t g = (size_t)blockIdx.x * blockDim.x + threadIdx.x;  // [0, B*C*R3)
    int v  = (int)(g % R3_);
    int bc = (int)(g / R3_);
    int b  = bc / C_;
    float cnt = counts[(size_t)b * R3_ + v];
    vox[g] = vox[g] / fmaxf(cnt, 1.0f);
}

extern "C" void kernel_launch(void* const* d_in, const int* in_sizes, int n_in,
                              void* d_out, int out_size, void* d_ws, size_t ws_size,
                              hipStream_t stream) {
    (void)in_sizes; (void)n_in; (void)out_size; (void)ws_size;
    const float* features = (const float*)d_in[0];   // (B, N, C) f32
    const float* coords   = (const float*)d_in[1];   // (B, N, 3) f32

    float* out = (float*)d_out;
    float* vox = out;                                 // (B, C, R3)
    float* nc  = out + (size_t)B_ * C_ * R3_;         // (B, N, 3)

    float* sumsq  = (float*)d_ws;                     // 24 floats (256B slot)
    float* counts = (float*)((char*)d_ws + 256);      // B*R3 floats (1 MB)

    hipMemsetAsync(d_ws, 0, 256 + (size_t)B_ * R3_ * sizeof(float), stream);
    hipMemsetAsync(d_out, 0, (size_t)B_ * C_ * R3_ * sizeof(float), stream);

    k_sumsq  <<<(B_ * N_) / TPB,   TPB, 0, stream>>>(coords, sumsq);
    k_voxelize<<<(B_ * N_) / PTILE, TPB, 0, stream>>>(features, coords, sumsq,
                                                      vox, nc, counts);
    k_div    <<<(size_t)B_ * C_ * R3_ / TPB, TPB, 0, stream>>>(vox, counts);
}